// SpeckleDistribution_75926431858947
// MI455X (gfx1250) — compile-verified
//
#include <hip/hip_runtime.h>

typedef __attribute__((ext_vector_type(2))) float v2f;
typedef __attribute__((ext_vector_type(8))) float v8f;

#define TSZ 50          // logical matrix size
#define NP 64           // padded size
#define LDSS 65         // LDS row stride (padded: conflict-free column reads)
#define NTHREADS 128    // 4 wave32s
#define SQUARINGS 6     // scale by 2^6 -> ||X|| <= ~0.4
#define TAYLOR 8        // Taylor degree after scaling

// D = (A*B)*invk  (+ I if addI), 64x64 f32 matmul via V_WMMA_F32_16X16X4_F32.
// Wave w computes output rows [16w,16w+16). A-fragments hoisted across the
// 4 column tiles; B-fragments bulk-preloaded per tile so DS loads clause up;
// two independent accumulators break the WMMA RAW chain.
__device__ __forceinline__ void mm64(const float* __restrict__ A,
                                     const float* __restrict__ B,
                                     float* __restrict__ D,
                                     float invk, int addI,
                                     int wave, int lane) {
  const int hf   = lane >> 4;       // half of the wave
  const int lrow = lane & 15;
  const int arow = wave * 16 + lrow;

  // A 16x4 fragment layout (32-bit): lanes 0-15 row M=lane, v0=K0,v1=K1;
  // lanes 16-31 row M=lane-16, v0=K2,v1=K3.
  v2f afrag[16];
#pragma unroll
  for (int ks = 0; ks < 16; ++ks) {
    const int k = ks * 4 + 2 * hf;
    afrag[ks].x = A[arow * LDSS + k];
    afrag[ks].y = A[arow * LDSS + k + 1];
  }

#pragma unroll
  for (int tc = 0; tc < 4; ++tc) {
    const int col = tc * 16 + lrow;

    // Bulk-preload all B 4x16 fragments for this column tile.
    v2f bfrag[16];
#pragma unroll
    for (int ks = 0; ks < 16; ++ks) {
      const int k = ks * 4 + 2 * hf;
      bfrag[ks].x = B[k * LDSS + col];
      bfrag[ks].y = B[(k + 1) * LDSS + col];
    }

    v8f c0 = {0.f, 0.f, 0.f, 0.f, 0.f, 0.f, 0.f, 0.f};
    v8f c1 = {0.f, 0.f, 0.f, 0.f, 0.f, 0.f, 0.f, 0.f};
#pragma unroll
    for (int ks = 0; ks < 16; ks += 2) {
      c0 = __builtin_amdgcn_wmma_f32_16x16x4_f32(
          false, afrag[ks],     false, bfrag[ks],     (short)0, c0, false, false);
      c1 = __builtin_amdgcn_wmma_f32_16x16x4_f32(
          false, afrag[ks + 1], false, bfrag[ks + 1], (short)0, c1, false, false);
    }

    // C/D layout: VGPR i -> row M=i (lanes 0-15) / M=i+8 (lanes 16-31).
#pragma unroll
    for (int i = 0; i < 8; ++i) {
      const int row = wave * 16 + i + 8 * hf;
      float v = (c0[i] + c1[i]) * invk;
      if (addI && row == col) v += 1.0f;
      D[row * LDSS + col] = v;
    }
  }
}

__global__ void __launch_bounds__(NTHREADS)
speckle_expm_kernel(const float* __restrict__ x,
                    const float* __restrict__ lmbda_sqrt,
                    float* __restrict__ out) {
  __shared__ float Xs[NP * LDSS];
  __shared__ float Ra[NP * LDSS];
  __shared__ float Rb[NP * LDSS];
  __shared__ float lam[NP];
  __shared__ float rk[TAYLOR];      // reciprocal table: kills the VALU divide

  const int pid  = blockIdx.x;      // problem id in [0, B*P)
  const int tid  = threadIdx.x;
  const int lane = tid & 31;
  const int wave = tid >> 5;

  if (tid < NP) {
    const float ls = (tid < TSZ) ? lmbda_sqrt[tid] : 0.0f;
    lam[tid] = ls * ls;
  }
  if (tid < TAYLOR) rk[tid] = 1.0f / (float)(tid == 0 ? 1 : tid);
  __syncthreads();

  const float s = x[pid];
  const float t = s * (1.0f / (float)(1 << SQUARINGS));  // s / 64

  // Build X = t*theta (zero-padded to 64x64; padding rows give an identity
  // block under expm, irrelevant to element [0,49]).
  // Seed Horner: R = I + X/TAYLOR  (= R_{TAYLOR-1}, one matmul for free).
  for (int idx = tid; idx < NP * NP; idx += NTHREADS) {
    const int row = idx >> 6;
    const int col = idx & 63;
    float v = 0.0f;
    if (row < TSZ) {
      if (col == row)                           v = -lam[row] * t;
      else if (col == row + 1 && row < TSZ - 1) v =  lam[row] * t;
    }
    Xs[row * LDSS + col] = v;
    float r = v * (1.0f / (float)TAYLOR);
    if (row == col) r += 1.0f;
    Ra[row * LDSS + col] = r;
  }
  __syncthreads();

  float* Rc = Ra;
  float* Rn = Rb;

  // Horner: R_{k-1} = I + (X/k) * R_k, k = 7..1  -> R_0 = Taylor-8 of exp(X)
  for (int k = TAYLOR - 1; k >= 1; --k) {
    mm64(Xs, Rc, Rn, rk[k], 1, wave, lane);
    __syncthreads();
    float* sw = Rc; Rc = Rn; Rn = sw;
  }

  // Squarings: R <- R*R, 6 times -> expm(s*theta)
  for (int sq = 0; sq < SQUARINGS; ++sq) {
    mm64(Rc, Rc, Rn, 1.0f, 0, wave, lane);
    __syncthreads();
    float* sw = Rc; Rc = Rn; Rn = sw;
  }

  // dist = lambda_{T-1} * E[0, T-1]   (theta @ ones has a single nonzero)
  if (tid == 0) out[pid] = lam[TSZ - 1] * Rc[0 * LDSS + (TSZ - 1)];
}

extern "C" void kernel_launch(void* const* d_in, const int* in_sizes, int n_in,
                              void* d_out, int out_size, void* d_ws, size_t ws_size,
                              hipStream_t stream) {
  const float* x  = (const float*)d_in[0];        // [64, 256] f32
  const float* ls = (const float*)d_in[1];        // [50] f32
  float* out = (float*)d_out;                     // [64, 256] f32
  const int nprob = in_sizes[0];                  // 16384
  speckle_expm_kernel<<<nprob, NTHREADS, 0, stream>>>(x, ls, out);
  (void)n_in; (void)out_size; (void)d_ws; (void)ws_size;
}